// PowerDense_2465311228654
// MI455X (gfx1250) — compile-verified
//
#include <hip/hip_runtime.h>
#include <stdint.h>

namespace {

constexpr int kB       = 8192;
constexpr int kK       = 128;
constexpr int kO       = 128;
constexpr int kBT      = 32;    // batch rows per block
constexpr int kThreads = 256;   // 8 wave32 waves
constexpr int kLPad    = 4;     // LDS row padding for the log tile

typedef unsigned int u32x4 __attribute__((ext_vector_type(4)));
typedef int          i32x8 __attribute__((ext_vector_type(8)));
typedef int          i32x4 __attribute__((ext_vector_type(4)));

__device__ __forceinline__ float fast_exp2(float v) { return __builtin_amdgcn_exp2f(v); }
__device__ __forceinline__ float fast_log2(float v) { return __builtin_amdgcn_logf(v); }
__device__ __forceinline__ float fast_sigmoid(float v) {
  // 1 / (1 + 2^(-v*log2(e)))
  return __builtin_amdgcn_rcpf(1.0f + fast_exp2(-1.44269504088896340736f * v));
}

__global__ __launch_bounds__(kThreads)
void powerdense_kernel(const float* __restrict__ x,
                       const float* __restrict__ w,
                       const float* __restrict__ bias,
                       float* __restrict__ out) {
  __shared__ float sw[kK][kO];            // 64 KB: weight matrix, TDM-loaded
  __shared__ float sl[kBT][kK + kLPad];   // 16.5 KB: log2(x) tile

  const int tid = threadIdx.x;
  const int b0  = blockIdx.x * kBT;

  // ---- Wave 0 kicks off one TDM DMA: w[128][128] f32 (64 KB) -> LDS ----
  if (tid < 32) {
    const uint64_t ga      = (uint64_t)(uintptr_t)w;
    const uint32_t lds_off = (uint32_t)(uintptr_t)(&sw[0][0]);  // low 32 bits = LDS offset
    u32x4 g0;
    g0[0] = 1u;                                            // count=1 valid descriptor
    g0[1] = lds_off;                                       // lds_addr
    g0[2] = (uint32_t)(ga & 0xffffffffu);                  // global_addr[31:0]
    g0[3] = (uint32_t)((ga >> 32) & 0x01ffffffu)           // global_addr[56:32]
          | (2u << 30);                                    // type=2 ("image")
    i32x8 g1;
    g1[0] = (2 << 16);       // data_size=2 -> 4 bytes; no multicast/iterate/pad
    g1[1] = (kO << 16);      // tensor_dim0[15:0] = 128 (row length)
    g1[2] = (kK << 16);      // tensor_dim1[15:0] = 128 (rows)
    g1[3] = (kO << 16);      // tile_dim0 = 128
    g1[4] = kK;              // tile_dim1 = 128, tile_dim2 = 0
    g1[5] = kO;              // tensor_dim0_stride = 128 (low 32)
    g1[6] = 0;
    g1[7] = 0;
    i32x4 g2 = {0, 0, 0, 0};
    i32x4 g3 = {0, 0, 0, 0};
    i32x8 g4 = {0, 0, 0, 0, 0, 0, 0, 0};  // extra operand on 6-arg toolchain
    __builtin_amdgcn_tensor_load_to_lds(g0, g1, g2, g3, g4, 0);
  }

  // ---- All threads: load x tile (32x128), take log2 once, stage in LDS ----
  // (overlaps with the TDM transfer above)
  #pragma unroll
  for (int g = 0; g < (kBT * kK) / (4 * kThreads); ++g) {  // 4 float4 per thread
    const int e = 4 * (tid + g * kThreads);
    const int r = e >> 7;           // /128
    const int c = e & (kK - 1);
    const float4 v = *reinterpret_cast<const float4*>(&x[(size_t)(b0 + r) * kK + c]);
    float4 lv;
    lv.x = fast_log2(v.x);
    lv.y = fast_log2(v.y);
    lv.z = fast_log2(v.z);
    lv.w = fast_log2(v.w);
    *reinterpret_cast<float4*>(&sl[r][c]) = lv;
  }

  if (tid < 32) __builtin_amdgcn_s_wait_tensorcnt(0);  // issuing wave drains TDM
  __syncthreads();                                     // publish sw + sl to all waves

  // ---- Main loop: thread owns (1 row b, 16 cols o); 2048 v_exp per thread ----
  const int br = tid >> 3;          // 0..31
  const int og = (tid & 7) << 4;    // 0,16,...,112

  float acc[16];
  #pragma unroll
  for (int i = 0; i < 16; ++i) acc[i] = 0.0f;

  #pragma unroll 2
  for (int k = 0; k < kK; ++k) {
    const float lb = sl[br][k];     // broadcast across the 8 threads of this row
    #pragma unroll
    for (int j = 0; j < 4; ++j) {
      const float4 wv = *reinterpret_cast<const float4*>(&sw[k][og + 4 * j]);
      acc[4 * j + 0] += fast_exp2(wv.x * lb);
      acc[4 * j + 1] += fast_exp2(wv.y * lb);
      acc[4 * j + 2] += fast_exp2(wv.z * lb);
      acc[4 * j + 3] += fast_exp2(wv.w * lb);
    }
  }

  // ---- Epilogue: + bias, sigmoid, coalesced float4 stores ----
  float* orow = out + (size_t)(b0 + br) * kO + og;
  #pragma unroll
  for (int j = 0; j < 4; ++j) {
    const float4 bv = *reinterpret_cast<const float4*>(&bias[og + 4 * j]);
    float4 r;
    r.x = fast_sigmoid(acc[4 * j + 0] + bv.x);
    r.y = fast_sigmoid(acc[4 * j + 1] + bv.y);
    r.z = fast_sigmoid(acc[4 * j + 2] + bv.z);
    r.w = fast_sigmoid(acc[4 * j + 3] + bv.w);
    *reinterpret_cast<float4*>(&orow[4 * j]) = r;
  }
}

}  // namespace

extern "C" void kernel_launch(void* const* d_in, const int* in_sizes, int n_in,
                              void* d_out, int out_size, void* d_ws, size_t ws_size,
                              hipStream_t stream) {
  const float* x    = (const float*)d_in[0];
  const float* w    = (const float*)d_in[1];
  const float* bias = (const float*)d_in[2];
  float*       out  = (float*)d_out;

  dim3 grid(kB / kBT);   // 256 blocks
  dim3 block(kThreads);  // 256 threads = 8 wave32
  hipLaunchKernelGGL(powerdense_kernel, grid, block, 0, stream, x, w, bias, out);
}